// GasSplattingModel_5995774345889
// MI455X (gfx1250) — compile-verified
//
#include <hip/hip_runtime.h>
#include <math.h>

typedef __attribute__((ext_vector_type(2))) float v2f;
typedef __attribute__((ext_vector_type(8))) float v8f;

#if defined(__HIP_DEVICE_COMPILE__)
#if !__has_builtin(__builtin_amdgcn_wmma_f32_16x16x4_f32)
#error "wmma_f32_16x16x4_f32 builtin not available on device target"
#endif
#endif

#define MAP_SIZE   100.0f
#define EPS_SCALE  1e-07f
#define SQRT_2PI   2.5066282746310002f
#define NEG_HALF_LOG2E (-0.7213475204444817f)  /* -0.5 * log2(e) */

static __device__ __forceinline__ v8f wmma4(v2f a, v2f b, v8f c) {
#if defined(__HIP_DEVICE_COMPILE__)
    // D = A(16x4,f32) * B(4x16,f32) + C(16x16,f32)
    return __builtin_amdgcn_wmma_f32_16x16x4_f32(
        /*neg_a=*/false, a, /*neg_b=*/false, b,
        /*c_mod=*/(short)0, c, /*reuse_a=*/false, /*reuse_b=*/false);
#else
    return c;  // host pass stub; never executed
#endif
}

static __device__ __forceinline__ float fast_rsq(float x) {
#if defined(__HIP_DEVICE_COMPILE__) && __has_builtin(__builtin_amdgcn_rsqf)
    return __builtin_amdgcn_rsqf(x);
#else
    return __frsqrt_rn(x);
#endif
}

static __device__ __forceinline__ float fast_exp2(float x) {
#if defined(__HIP_DEVICE_COMPILE__) && __has_builtin(__builtin_amdgcn_exp2f)
    return __builtin_amdgcn_exp2f(x);
#else
    return exp2f(x);
#endif
}

// ---------------------------------------------------------------------------
// Kernel 1: per-gaussian feature precompute (K elements, trivial cost).
// ws float layout:
//   GA   @ 0      : K x float4 {m11, m12, m22, 0}
//   GB0  @ 4K     : K x float4 {t1, t2, -m11, -m12}
//   GB1  @ 8K     : K x float4 {-m22, 0, 0, 0}
//   GC0  @ 12K    : K x float4 {q, -2*t1, -2*t2, m11}
//   GC1  @ 16K    : K x float4 {2*m12, m22, 0, 0}
//   CONC @ 20K    : K x float  softplus(_concentration)
// ---------------------------------------------------------------------------
__global__ void gs_prep_kernel(const float* __restrict__ pos_,
                               const float* __restrict__ conc_,
                               const float* __restrict__ scale_,
                               const float* __restrict__ rot_,
                               float* __restrict__ ws, int K) {
    int k = blockIdx.x * blockDim.x + threadIdx.x;
    if (k >= K) return;

    float4* GA  = (float4*)(ws);
    float4* GB0 = (float4*)(ws + 4 * (size_t)K);
    float4* GB1 = (float4*)(ws + 8 * (size_t)K);
    float4* GC0 = (float4*)(ws + 12 * (size_t)K);
    float4* GC1 = (float4*)(ws + 16 * (size_t)K);
    float*  CC  = ws + 20 * (size_t)K;

    float px = MAP_SIZE / (1.0f + __expf(-pos_[2 * k + 0]));  // sigmoid * 100
    float py = MAP_SIZE / (1.0f + __expf(-pos_[2 * k + 1]));
    float e0 = __expf(scale_[2 * k + 0]);
    float e1 = __expf(scale_[2 * k + 1]);
    float si0 = 1.0f / (e0 * e0 + EPS_SCALE);
    float si1 = 1.0f / (e1 * e1 + EPS_SCALE);
    float sr, cr;
    sincosf(rot_[k], &sr, &cr);

    float m11 = cr * cr * si0 + sr * sr * si1;
    float m12 = cr * sr * (si0 - si1);
    float m22 = sr * sr * si0 + cr * cr * si1;
    float t1 = m11 * px + m12 * py;
    float t2 = m12 * px + m22 * py;
    float q  = t1 * px + t2 * py;  // m11*px^2 + 2*m12*px*py + m22*py^2

    GA[k]  = make_float4(m11, m12, m22, 0.0f);
    GB0[k] = make_float4(t1, t2, -m11, -m12);
    GB1[k] = make_float4(-m22, 0.0f, 0.0f, 0.0f);
    GC0[k] = make_float4(q, -2.0f * t1, -2.0f * t2, m11);
    GC1[k] = make_float4(2.0f * m12, m22, 0.0f, 0.0f);
    CC[k]  = log1pf(__expf(conc_[k]));  // softplus
}

// ---------------------------------------------------------------------------
// Kernel 2: one wave per 16 rays. Loop over gaussian tiles of 16; per tile
// compute a/b/c0 via 5x V_WMMA_F32_16X16X4_F32, then VALU epilogue
// (1 rsq + 1 exp2 per pair). Ray sums live in the 8 accumulator lanes of a
// v8f across the whole loop; one in-wave shuffle reduction at the end.
// C/D layout: VGPR j, lanes 0-15 -> (M=j,   N=lane),
//                     lanes16-31 -> (M=j+8, N=lane-16).
// ---------------------------------------------------------------------------
__global__ void __launch_bounds__(256)
gs_splat_kernel(const float* __restrict__ p_rays,
                const float* __restrict__ u_rays,
                const float* __restrict__ ws,
                float* __restrict__ out, int R, int K) {
    const int lane = threadIdx.x & 31;
    const int wave = threadIdx.x >> 5;
    const int rbase = (blockIdx.x * 8 + wave) * 16;

    const int m  = lane & 15;   // M (rays) for A; N (gaussians) for B
    const int hi = lane >> 4;   // 0: K-slots {0,1}, 1: K-slots {2,3}
    const int kk = hi * 2;

    const float4* GA  = (const float4*)(ws);
    const float4* GB0 = (const float4*)(ws + 4 * (size_t)K);
    const float4* GB1 = (const float4*)(ws + 8 * (size_t)K);
    const float4* GC0 = (const float4*)(ws + 12 * (size_t)K);
    const float4* GC1 = (const float4*)(ws + 16 * (size_t)K);
    const float*  CC  = ws + 20 * (size_t)K;

    // ---- per-ray features (invariant over the gaussian loop) ----
    int r = rbase + m;
    if (r >= R) r = R - 1;  // clamp reads; writes guarded below
    const float p1 = p_rays[2 * r + 0];
    const float p2 = p_rays[2 * r + 1];
    const float u1 = u_rays[2 * r + 0];
    const float u2 = u_rays[2 * r + 1];

    const float RA[4]  = {u1 * u1, 2.0f * u1 * u2, u2 * u2, 0.0f};
    const float RB0[4] = {u1, u2, u1 * p1, u1 * p2 + u2 * p1};
    const float RB1[4] = {u2 * p2, 0.0f, 0.0f, 0.0f};
    const float RC0[4] = {1.0f, p1, p2, p1 * p1};
    const float RC1[4] = {p1 * p2, p2 * p2, 0.0f, 0.0f};

    // A operand regs: VGPR0 = K=kk (this half), VGPR1 = K=kk+1
    v2f aA, aB0, aB1, aC0, aC1;
    aA[0]  = RA[kk];  aA[1]  = RA[kk + 1];
    aB0[0] = RB0[kk]; aB0[1] = RB0[kk + 1];
    aB1[0] = RB1[kk]; aB1[1] = RB1[kk + 1];
    aC0[0] = RC0[kk]; aC0[1] = RC0[kk + 1];
    aC1[0] = RC1[kk]; aC1[1] = RC1[kk + 1];

    v8f sum = {};
    const v8f zero = {};

    const int ntiles = K >> 4;
    for (int gt = 0; gt < ntiles; ++gt) {
        const int g = (gt << 4) + m;  // gaussian column N for this lane

        const float4 ga  = GA[g];
        const float4 gb0 = GB0[g];
        const float4 gb1 = GB1[g];
        const float4 gc0 = GC0[g];
        const float4 gc1 = GC1[g];
        const float  cc  = CC[g];

        // B operand regs: VGPR0 = row K=kk, VGPR1 = row K=kk+1, col N=m
        v2f bA, bB0, bB1, bC0, bC1;
        bA[0]  = hi ? ga.z  : ga.x;   bA[1]  = hi ? ga.w  : ga.y;
        bB0[0] = hi ? gb0.z : gb0.x;  bB0[1] = hi ? gb0.w : gb0.y;
        bB1[0] = hi ? gb1.z : gb1.x;  bB1[1] = hi ? gb1.w : gb1.y;
        bC0[0] = hi ? gc0.z : gc0.x;  bC0[1] = hi ? gc0.w : gc0.y;
        bC1[0] = hi ? gc1.z : gc1.x;  bC1[1] = hi ? gc1.w : gc1.y;

        // 5 WMMAs -> 16x16 tiles of a, b, c0
        v8f acc_a = wmma4(aA, bA, zero);
        v8f acc_b = wmma4(aB0, bB0, zero);
        acc_b     = wmma4(aB1, bB1, acc_b);
        v8f acc_c = wmma4(aC0, bC0, zero);
        acc_c     = wmma4(aC1, bC1, acc_c);

        // epilogue: conc * sqrt(2pi)*rsqrt(a) * exp(-0.5*(c0 - (b*rsqrt(a))^2))
#pragma unroll
        for (int j = 0; j < 8; ++j) {
            float a    = acc_a[j] + 1e-8f;
            float rsqa = fast_rsq(a);
            float br   = acc_b[j] * rsqa;
            float t    = acc_c[j] - br * br;
            sum[j] += cc * (SQRT_2PI * rsqa) * fast_exp2(NEG_HALF_LOG2E * t);
        }
    }

    // Reduce over the 16 gaussian columns held across each 16-lane half.
#pragma unroll
    for (int j = 0; j < 8; ++j) {
        float v = sum[j];
        v += __shfl_xor(v, 1, 32);
        v += __shfl_xor(v, 2, 32);
        v += __shfl_xor(v, 4, 32);
        v += __shfl_xor(v, 8, 32);
        const int ray = rbase + hi * 8 + j;  // lanes0-15: M=j, lanes16-31: M=j+8
        if (m == 0 && ray < R) out[ray] = v;
    }
}

extern "C" void kernel_launch(void* const* d_in, const int* in_sizes, int n_in,
                              void* d_out, int out_size, void* d_ws, size_t ws_size,
                              hipStream_t stream) {
    const float* pos_   = (const float*)d_in[0];  // (K,2)
    const float* conc_  = (const float*)d_in[1];  // (K,)
    const float* scale_ = (const float*)d_in[2];  // (K,2)
    const float* rot_   = (const float*)d_in[3];  // (K,)
    const float* p_rays = (const float*)d_in[4];  // (R,2)
    const float* u_rays = (const float*)d_in[5];  // (R,2)
    float* out = (float*)d_out;
    float* ws  = (float*)d_ws;

    const int K = in_sizes[1];
    const int R = in_sizes[4] / 2;

    // 1) per-gaussian features (needs 21*K floats of workspace = 84KB @ K=1024)
    gs_prep_kernel<<<(K + 255) / 256, 256, 0, stream>>>(pos_, conc_, scale_, rot_, ws, K);

    // 2) WMMA splat: one wave (32 lanes) per 16 rays, 8 waves per block
    const int ray_groups = (R + 15) / 16;          // 2048 @ R=32768
    const int blocks     = (ray_groups + 7) / 8;   // 256 blocks x 256 threads
    gs_splat_kernel<<<blocks, 256, 0, stream>>>(p_rays, u_rays, ws, out, R, K);
}